// DynaFlow_65695819760083
// MI455X (gfx1250) — compile-verified
//
#include <hip/hip_runtime.h>
#include <hip/hip_bf16.h>

// ---- problem constants ----
#define Tt 96
#define Bb 1024
#define Kk 10
#define Cc 16
#define Hh 32
#define Gg 256
#define Ll 4

// phase-1 tiling
#define BT   16            // graphs per block
#define RPB  (BT*Kk)       // 160 rows
#define NMT  (RPB/16)      // 10 M-tiles
#define NW1  4             // waves per block (128 threads)

typedef float v2f __attribute__((ext_vector_type(2)));
typedef float v8f __attribute__((ext_vector_type(8)));

static __device__ __forceinline__ v8f wmma4(v2f a, v2f b, v8f c) {
  // D = A(16x4 f32) * B(4x16 f32) + C(16x16 f32)
  return __builtin_amdgcn_wmma_f32_16x16x4_f32(false, a, false, b, (short)0, c, false, false);
}

static __device__ __forceinline__ v2f mk2(float a, float b) { v2f v; v.x = a; v.y = b; return v; }

static __device__ __forceinline__ v8f bcast8(float x) {
  v8f c;
#pragma unroll
  for (int i = 0; i < 8; ++i) c[i] = x;
  return c;
}

static __device__ __forceinline__ float sigm(float x) { return 1.0f / (1.0f + __expf(-x)); }

#if __has_builtin(__builtin_amdgcn_global_load_async_to_lds_b128)
#define HAVE_ASYNC_LDS 1
typedef int vsi4 __attribute__((vector_size(16)));
typedef __attribute__((address_space(1))) vsi4 g_vsi4;   // global (device) AS
typedef __attribute__((address_space(3))) vsi4 l_vsi4;   // LDS AS
static __device__ __forceinline__ void async_cp16(const float* src, float* dst) {
  __builtin_amdgcn_global_load_async_to_lds_b128((g_vsi4*)src, (l_vsi4*)dst, 0, 0);
}
#else
#define HAVE_ASYNC_LDS 0
#endif

// =====================================================================
// Phase 1: per-timestep GNN stack -> gi[t,b,:]   (all t parallel)
// =====================================================================
__global__ __launch_bounds__(128)
void gnn_kernel(const float* __restrict__ x,   const float* __restrict__ adj,
                const float* __restrict__ Wfi, const float* __restrict__ bfi,
                const float* __restrict__ Wz,  const float* __restrict__ Uz, const float* __restrict__ bz,
                const float* __restrict__ Wr,  const float* __restrict__ Ur, const float* __restrict__ br,
                const float* __restrict__ Wh,  const float* __restrict__ Uh, const float* __restrict__ bh,
                const float* __restrict__ Wgp, const float* __restrict__ bgp,
                float* __restrict__ gi)
{
  __shared__ float sh_h [RPB*Hh];
  __shared__ float sh_h0[RPB*Hh];
  __shared__ float sh_m [RPB*Hh];   // also holds n after gate-n GEMM
  __shared__ float sh_z [RPB*Hh];
  __shared__ float sh_r [RPB*Hh];   // r, then r*h
  __shared__ float sh_adj[BT*Kk*Kk];
  __shared__ float sh_w [6*Hh*Hh];  // Wz,Uz,Wr,Ur,Wh,Uh for current layer

  const int tid  = threadIdx.x;
  const int lane = tid & 31;
  const int wave = tid >> 5;
  const int ln   = lane & 15;   // row-in-Mtile (A) / col-in-Ntile (B,C,D)
  const int g    = lane >> 4;   // half-wave selector
  const int b0   = blockIdx.x * BT;
  const int t    = blockIdx.y;

  // ---- adjacency: add self loops, row-normalize ----
  for (int r = tid; r < BT*Kk; r += 128) {
    const int bl = r / Kk, i = r % Kk;
    const float* arow = adj + ((((size_t)t*Bb) + b0 + bl)*Kk + i)*Kk;
    float tmp[Kk]; float s = 0.f;
#pragma unroll
    for (int j = 0; j < Kk; ++j) { float v = arow[j] + (i == j ? 1.f : 0.f); tmp[j] = v; s += v; }
    const float inv = 1.f / (s + 1e-6f);
#pragma unroll
    for (int j = 0; j < Kk; ++j) sh_adj[r*Kk + j] = tmp[j] * inv;
  }

  // ---- h0 = x_t @ W_fi + b_fi  (WMMA, K=16 -> 4 chained steps) ----
  for (int mt = wave; mt < NMT; mt += NW1) {
    const int arow = mt*16 + ln;            // 0..159
    const int bl = arow / Kk, kk = arow % Kk;
    const float* xr = x + ((((size_t)t*Bb) + b0 + bl)*Kk + kk)*Cc;
#pragma unroll
    for (int nt = 0; nt < 2; ++nt) {
      const int col = nt*16 + ln;
      v8f c = bcast8(bfi[col]);
#pragma unroll
      for (int kb = 0; kb < Cc; kb += 4) {
        v2f a = mk2(xr[kb + 2*g], xr[kb + 2*g + 1]);
        v2f b = mk2(Wfi[(kb + 2*g)*Hh + col], Wfi[(kb + 2*g + 1)*Hh + col]);
        c = wmma4(a, b, c);
      }
#pragma unroll
      for (int i = 0; i < 8; ++i) {
        const int rr = mt*16 + 8*g + i;
        sh_h0[rr*Hh + col] = c[i];
        sh_h [rr*Hh + col] = c[i];
      }
    }
  }

  // ---- L stacked GNN-GRU layers ----
  for (int l = 0; l < Ll; ++l) {
    __syncthreads();
    {
      const float* srcs[6] = { Wz + l*Hh*Hh, Uz + l*Hh*Hh, Wr + l*Hh*Hh,
                               Ur + l*Hh*Hh, Wh + l*Hh*Hh, Uh + l*Hh*Hh };
      for (int i = tid; i < 6*Hh*Hh; i += 128) sh_w[i] = srcs[i >> 10][i & 1023];
    }
    __syncthreads();

    // m = adj_norm @ h  (tiny 10x10, VALU)
    for (int e = tid; e < RPB*Hh; e += 128) {
      const int r = e >> 5, col = e & 31;
      const int bl = r / Kk, kk = r % Kk;
      const float* arow = &sh_adj[(bl*Kk + kk)*Kk];
      const float* hcol = &sh_h[bl*Kk*Hh + col];
      float s = 0.f;
#pragma unroll
      for (int j = 0; j < Kk; ++j) s += arow[j] * hcol[j*Hh];
      sh_m[e] = s;
    }
    __syncthreads();

    // z = sigmoid(m@Wz + h@Uz + bz) ; r = sigmoid(m@Wr + h@Ur + br)
    for (int mt = wave; mt < NMT; mt += NW1) {
      const int arow = mt*16 + ln;
#pragma unroll
      for (int nt = 0; nt < 2; ++nt) {
        const int col = nt*16 + ln;
        v8f cz = bcast8(bz[l*Hh + col]);
        v8f cr = bcast8(br[l*Hh + col]);
#pragma unroll
        for (int kb = 0; kb < Hh; kb += 4) {
          v2f am = mk2(sh_m[arow*Hh + kb + 2*g], sh_m[arow*Hh + kb + 2*g + 1]);
          v2f ah = mk2(sh_h[arow*Hh + kb + 2*g], sh_h[arow*Hh + kb + 2*g + 1]);
          cz = wmma4(am, mk2(sh_w[0*1024 + (kb+2*g)*Hh + col], sh_w[0*1024 + (kb+2*g+1)*Hh + col]), cz);
          cz = wmma4(ah, mk2(sh_w[1*1024 + (kb+2*g)*Hh + col], sh_w[1*1024 + (kb+2*g+1)*Hh + col]), cz);
          cr = wmma4(am, mk2(sh_w[2*1024 + (kb+2*g)*Hh + col], sh_w[2*1024 + (kb+2*g+1)*Hh + col]), cr);
          cr = wmma4(ah, mk2(sh_w[3*1024 + (kb+2*g)*Hh + col], sh_w[3*1024 + (kb+2*g+1)*Hh + col]), cr);
        }
#pragma unroll
        for (int i = 0; i < 8; ++i) {
          const int rr = mt*16 + 8*g + i;
          sh_z[rr*Hh + col] = sigm(cz[i]);
          sh_r[rr*Hh + col] = sigm(cr[i]);
        }
      }
    }
    __syncthreads();

    // rh = r * h  (overwrite r)
    for (int e = tid; e < RPB*Hh; e += 128) sh_r[e] = sh_r[e] * sh_h[e];
    __syncthreads();

    // n = tanh(m@Wh + rh@Uh + bh) -> store into sh_m (each wave writes only its own rows)
    for (int mt = wave; mt < NMT; mt += NW1) {
      const int arow = mt*16 + ln;
      v8f cn[2];
#pragma unroll
      for (int nt = 0; nt < 2; ++nt) {
        const int col = nt*16 + ln;
        cn[nt] = bcast8(bh[l*Hh + col]);
#pragma unroll
        for (int kb = 0; kb < Hh; kb += 4) {
          v2f am = mk2(sh_m[arow*Hh + kb + 2*g], sh_m[arow*Hh + kb + 2*g + 1]);
          v2f ar = mk2(sh_r[arow*Hh + kb + 2*g], sh_r[arow*Hh + kb + 2*g + 1]);
          cn[nt] = wmma4(am, mk2(sh_w[4*1024 + (kb+2*g)*Hh + col], sh_w[4*1024 + (kb+2*g+1)*Hh + col]), cn[nt]);
          cn[nt] = wmma4(ar, mk2(sh_w[5*1024 + (kb+2*g)*Hh + col], sh_w[5*1024 + (kb+2*g+1)*Hh + col]), cn[nt]);
        }
      }
#pragma unroll
      for (int nt = 0; nt < 2; ++nt) {
        const int col = nt*16 + ln;
#pragma unroll
        for (int i = 0; i < 8; ++i) {
          const int rr = mt*16 + 8*g + i;
          sh_m[rr*Hh + col] = tanhf(cn[nt][i]);
        }
      }
    }
    __syncthreads();

    // h = (1-z)*h + z*n + h0
    for (int e = tid; e < RPB*Hh; e += 128) {
      const float zz = sh_z[e];
      sh_h[e] = (1.f - zz) * sh_h[e] + zz * sh_m[e] + sh_h0[e];
    }
  }
  __syncthreads();

  // ---- gi = comb @ Wgp + bgp ; comb row b is contiguous 320 floats of sh_h ----
  for (int nt = wave*4; nt < wave*4 + 4; ++nt) {   // 4 waves x 4 N-tiles = 16
    const int col = nt*16 + ln;
    v8f c = bcast8(bgp[col]);
#pragma unroll 8
    for (int kb = 0; kb < Kk*Hh; kb += 4) {
      v2f a = mk2(sh_h[ln*(Kk*Hh) + kb + 2*g], sh_h[ln*(Kk*Hh) + kb + 2*g + 1]);
      v2f b = mk2(Wgp[(kb + 2*g)*Gg + col],    Wgp[(kb + 2*g + 1)*Gg + col]);
      c = wmma4(a, b, c);
    }
#pragma unroll
    for (int i = 0; i < 8; ++i) {
      const int bb = 8*g + i;
      gi[(((size_t)t*Bb) + b0 + bb)*Gg + col] = c[i];
    }
  }
}

// =====================================================================
// Phase 2: full 96-step GARU recurrence in ONE launch.
// Row-independence: h_new[row] depends only on h[row], gi[t,row], so each
// block owns 16 rows for all timesteps; h/z/r live in LDS; gi tiles are
// double-buffered and prefetched via async global->LDS (ASYNCcnt path).
// 64 blocks x 16 waves; wave w owns N-tile w (cols w*16..w*16+15).
// =====================================================================
__global__ __launch_bounds__(512)
void garu_seq_kernel(const float* __restrict__ gi,
                     const float* __restrict__ Wgz, const float* __restrict__ Ugz, const float* __restrict__ bgz,
                     const float* __restrict__ Wgr, const float* __restrict__ Ugr, const float* __restrict__ bgr,
                     const float* __restrict__ Wgn, const float* __restrict__ Ugn, const float* __restrict__ bgn,
                     float* __restrict__ hout)
{
  __shared__ float sh_h [16*Gg];
  __shared__ float sh_z [16*Gg];
  __shared__ float sh_r [16*Gg];      // r, then r*h
  __shared__ float sh_gi[2][16*Gg];   // double-buffered gi tile

  const int tid  = threadIdx.x;
  const int lane = tid & 31;
  const int wave = tid >> 5;    // 0..15 == N-tile
  const int ln   = lane & 15;
  const int g    = lane >> 4;
  const int row0 = blockIdx.x * 16;
  const int col  = wave*16 + ln;

  // h starts at zero
  for (int i = tid; i < 16*Gg; i += 512) sh_h[i] = 0.f;

  // stage gi tile for timestep t into buffer buf
  auto stage = [&](int buf, int t) {
    const float* src = gi + (((size_t)t*Bb) + row0)*Gg;
    float* dst = sh_gi[buf];
#if HAVE_ASYNC_LDS
#pragma unroll
    for (int i = tid*4; i < 16*Gg; i += 512*4) {
      async_cp16(src + i, dst + i);
    }
#else
    const float4* s4 = (const float4*)src;
    float4* d4 = (float4*)dst;
#pragma unroll
    for (int i = tid; i < (16*Gg)/4; i += 512) d4[i] = s4[i];
#endif
  };

  stage(0, 0);

  for (int t = 0; t < Tt; ++t) {
    const int buf = t & 1;
#if HAVE_ASYNC_LDS
    asm volatile("s_wait_asynccnt 0x0" ::: "memory");
#endif
    __syncthreads();                      // gi[t] ready; prev-step h update published
    if (t + 1 < Tt) stage(buf ^ 1, t + 1); // prefetch next tile while computing

    const float* giL = sh_gi[buf];

    // ---- z & r gates: [16,256]@[256,16] twice, fused A loads ----
    v8f cz = bcast8(bgz[col]);
    v8f cr = bcast8(bgr[col]);
#pragma unroll 8
    for (int kb = 0; kb < Gg; kb += 4) {
      v2f a1 = mk2(giL [ln*Gg + kb + 2*g], giL [ln*Gg + kb + 2*g + 1]);
      v2f a2 = mk2(sh_h[ln*Gg + kb + 2*g], sh_h[ln*Gg + kb + 2*g + 1]);
      cz = wmma4(a1, mk2(Wgz[(kb+2*g)*Gg + col], Wgz[(kb+2*g+1)*Gg + col]), cz);
      cz = wmma4(a2, mk2(Ugz[(kb+2*g)*Gg + col], Ugz[(kb+2*g+1)*Gg + col]), cz);
      cr = wmma4(a1, mk2(Wgr[(kb+2*g)*Gg + col], Wgr[(kb+2*g+1)*Gg + col]), cr);
      cr = wmma4(a2, mk2(Ugr[(kb+2*g)*Gg + col], Ugr[(kb+2*g+1)*Gg + col]), cr);
    }
#pragma unroll
    for (int i = 0; i < 8; ++i) {
      const int rr = 8*g + i;
      sh_z[rr*Gg + col] = sigm(cz[i]);
      sh_r[rr*Gg + col] = sigm(cr[i]);
    }
    __syncthreads();

    // rh = r * h
    for (int i = tid; i < 16*Gg; i += 512) sh_r[i] = sh_r[i] * sh_h[i];
    __syncthreads();

    // ---- n gate + state update ----
    v8f cn = bcast8(bgn[col]);
#pragma unroll 8
    for (int kb = 0; kb < Gg; kb += 4) {
      v2f a1 = mk2(giL [ln*Gg + kb + 2*g], giL [ln*Gg + kb + 2*g + 1]);
      v2f a2 = mk2(sh_r[ln*Gg + kb + 2*g], sh_r[ln*Gg + kb + 2*g + 1]);
      cn = wmma4(a1, mk2(Wgn[(kb+2*g)*Gg + col], Wgn[(kb+2*g+1)*Gg + col]), cn);
      cn = wmma4(a2, mk2(Ugn[(kb+2*g)*Gg + col], Ugn[(kb+2*g+1)*Gg + col]), cn);
    }
    // h_new = (1-gz)*gn + gz*h  (each wave owns exactly these cells)
#pragma unroll
    for (int i = 0; i < 8; ++i) {
      const int rr = 8*g + i;
      const int idx = rr*Gg + col;
      const float gn = tanhf(cn[i]);
      const float gz = sh_z[idx];
      sh_h[idx] = (1.f - gz) * gn + gz * sh_h[idx];
    }
    // loop-top __syncthreads publishes h for next step
  }

  __syncthreads();
  for (int i = tid; i < 16*Gg; i += 512) {
    hout[((size_t)row0 + (i >> 8))*Gg + (i & 255)] = sh_h[i];
  }
}

// =====================================================================
// Phase 3: classifier  sigmoid(relu(h@W1+b1)@W2+b2)  (34 MFLOP, VALU)
// =====================================================================
__global__ __launch_bounds__(256)
void cls_kernel(const float* __restrict__ h, const float* __restrict__ W1, const float* __restrict__ b1,
                const float* __restrict__ W2, const float* __restrict__ b2, float* __restrict__ out)
{
  const int row = blockIdx.x * 256 + threadIdx.x;
  if (row >= Bb) return;
  float acc = b2[0];
  for (int j = 0; j < 64; ++j) {
    float s = b1[j];
#pragma unroll 8
    for (int k = 0; k < Gg; ++k) s += h[(size_t)row*Gg + k] * W1[k*64 + j];
    s = fmaxf(s, 0.f);
    acc += s * W2[j];
  }
  out[row] = sigm(acc);
}

// =====================================================================
extern "C" void kernel_launch(void* const* d_in, const int* in_sizes, int n_in,
                              void* d_out, int out_size, void* d_ws, size_t ws_size,
                              hipStream_t stream) {
  (void)in_sizes; (void)n_in; (void)out_size; (void)ws_size;
  const float* x   = (const float*)d_in[0];
  const float* adj = (const float*)d_in[1];
  const float* Wfi = (const float*)d_in[2];
  const float* bfi = (const float*)d_in[3];
  const float* Wz  = (const float*)d_in[4];
  const float* Uz  = (const float*)d_in[5];
  const float* bz  = (const float*)d_in[6];
  const float* Wr  = (const float*)d_in[7];
  const float* Ur  = (const float*)d_in[8];
  const float* br  = (const float*)d_in[9];
  const float* Wh  = (const float*)d_in[10];
  const float* Uh  = (const float*)d_in[11];
  const float* bh  = (const float*)d_in[12];
  const float* Wgp = (const float*)d_in[13];
  const float* bgp = (const float*)d_in[14];
  const float* Wgz = (const float*)d_in[15];
  const float* Ugz = (const float*)d_in[16];
  const float* bgz = (const float*)d_in[17];
  const float* Wgr = (const float*)d_in[18];
  const float* Ugr = (const float*)d_in[19];
  const float* bgr = (const float*)d_in[20];
  const float* Wgn = (const float*)d_in[21];
  const float* Ugn = (const float*)d_in[22];
  const float* bgn = (const float*)d_in[23];
  const float* W1  = (const float*)d_in[24];
  const float* b1  = (const float*)d_in[25];
  const float* W2  = (const float*)d_in[26];
  const float* b2  = (const float*)d_in[27];

  float* ws   = (float*)d_ws;
  float* gi   = ws;                                   // T*B*G floats
  float* hfin = gi + (size_t)Tt*Bb*Gg;                // B*G floats

  gnn_kernel<<<dim3(Bb/BT, Tt), 128, 0, stream>>>(
      x, adj, Wfi, bfi, Wz, Uz, bz, Wr, Ur, br, Wh, Uh, bh, Wgp, bgp, gi);

  garu_seq_kernel<<<dim3(Bb/16), 512, 0, stream>>>(
      gi, Wgz, Ugz, bgz, Wgr, Ugr, bgr, Wgn, Ugn, bgn, hfin);

  cls_kernel<<<(Bb + 255)/256, 256, 0, stream>>>(hfin, W1, b1, W2, b2, (float*)d_out);
}